// GCNConv_18476949308096
// MI455X (gfx1250) — compile-verified
//
#include <hip/hip_runtime.h>

typedef __attribute__((ext_vector_type(2))) float v2f;
typedef __attribute__((ext_vector_type(4))) float v4f;
typedef __attribute__((ext_vector_type(8))) float v8f;

#define D_FEAT 128
#define N_TILES 8            // 128 / 16
// LDS layout: K-pair interleaved W.  Row kk holds {W[2kk][n], W[2kk+1][n]}
// as float2 at Wlds[kk*PITCH + 2n].  PITCH = 256 data floats + 32 pad so that
// PITCH mod 64 == 32: lanes 16-31 (row kk+1) land on the opposite 32 banks
// from lanes 0-15 -> conflict-free wave-wide ds_load_b64.
#define PITCH 288
#define W_LDS_BYTES (64 * PITCH * 4)   // 73,728 B (dynamic LDS; WGP has 320 KB)

// ---------------------------------------------------------------------------
// Kernel 1: support = X @ W via V_WMMA_F32_16X16X4_F32, W staged in LDS in
// WMMA-operand order so each B fragment is ONE aligned b64 LDS read.
// Per k-step: prefetch ALL 8 B pairs (one DS clause, one wait), then fire the
// 8 WMMAs back-to-back -- next iteration's loads overlap the WMMA burst.
// A-frag (16x4 f32): lanes 0-15 hold M=lane K={k0,k0+1}; lanes 16-31 K={k0+2,k0+3}.
// B-frag (4x16 f32): lane holds {W[k0][n], W[k0+1][n]} = the interleaved pair.
// C/D (16x16 f32): VGPR v -> M = v (lanes 0-15) / M = v+8 (lanes 16-31).
// ---------------------------------------------------------------------------
__global__ __launch_bounds__(256) void gcn_gemm_wmma_f32(
    const float* __restrict__ X,   // [N, 128]
    const float* __restrict__ W,   // [128, 128]
    float* __restrict__ S,         // [N, 128] support (workspace)
    int n_mtiles) {
  extern __shared__ float Wlds[];  // 64 x PITCH floats (pad cols never read)

  // Cooperative stage, vectorized: 2048 float4-groups / 256 threads = 8 each.
  // Group g: load one float4 from each row of K-pair kk, interleave, store as
  // two contiguous 16B-aligned ds_store_b128 (PITCH*4 = 1152 = 72*16).
  {
    const int t = threadIdx.x;
#pragma unroll
    for (int i = 0; i < 8; ++i) {
      const int g  = t + i * 256;       // float4-group 0..2047
      const int kk = g >> 5;            // K-pair row 0..63
      const int n4 = (g & 31) << 2;     // column 0,4,...,124
      v4f r0 = *(const v4f*)&W[(size_t)(2 * kk) * D_FEAT + n4];
      v4f r1 = *(const v4f*)&W[(size_t)(2 * kk + 1) * D_FEAT + n4];
      v4f lo = {r0.x, r1.x, r0.y, r1.y};
      v4f hi = {r0.z, r1.z, r0.w, r1.w};
      *(v4f*)&Wlds[kk * PITCH + 2 * n4]     = lo;
      *(v4f*)&Wlds[kk * PITCH + 2 * n4 + 4] = hi;
    }
  }
  __syncthreads();

  const int mt = __builtin_amdgcn_readfirstlane(
      (int)(blockIdx.x * 8u + (threadIdx.x >> 5)));
  if (mt >= n_mtiles) return;           // wave-uniform exit: EXEC all-1s below

  const int lane = threadIdx.x & 31;
  const int half = lane >> 4;           // 0: lanes 0-15, 1: lanes 16-31
  const int l15  = lane & 15;

  const float* __restrict__ xrow = X + (size_t)(mt * 16 + l15) * D_FEAT;

  v8f acc[N_TILES] = {};                // zero accumulators

  for (int kb = 0; kb < D_FEAT / 4; ++kb) {
    const int k0 = kb * 4 + half * 2;
    v2f a = *(const v2f*)(xrow + k0);   // global_load_b64, contiguous pair

    // B pairs for this wave-half: row kk = k0/2 = kb*2 + half.
    // b[nt] is byte offset nt*128 off one base -> DS clause of 2addr_b64s.
    const v2f* __restrict__ brow =
        (const v2f*)&Wlds[(kb * 2 + half) * PITCH + 2 * l15];

    v2f b[N_TILES];
#pragma unroll
    for (int nt = 0; nt < N_TILES; ++nt) {
      b[nt] = brow[nt * 16];            // {W[k0][col], W[k0+1][col]}
    }

#pragma unroll
    for (int nt = 0; nt < N_TILES; ++nt) {
      acc[nt] = __builtin_amdgcn_wmma_f32_16x16x4_f32(
          /*neg_a=*/false, a, /*neg_b=*/false, b[nt],
          /*c_mod=*/(short)0, acc[nt], /*reuse_a=*/false, /*reuse_b=*/false);
    }
  }

  // Store: VGPR v of acc[nt] -> row mt*16 + half*8 + v, col nt*16 + l15
  float* __restrict__ srow = S + (size_t)(mt * 16 + half * 8) * D_FEAT + l15;
#pragma unroll
  for (int nt = 0; nt < N_TILES; ++nt) {
#pragma unroll
    for (int v = 0; v < 8; ++v) {
      srow[(size_t)v * D_FEAT + nt * 16] = acc[nt][v];
    }
  }
}

// ---------------------------------------------------------------------------
// Kernel 2: out[i, j] = bias[j]  (d_out is poisoned by the harness)
// ---------------------------------------------------------------------------
__global__ __launch_bounds__(256) void gcn_init_out(
    float* __restrict__ out, const float* __restrict__ bias, int n_elems) {
  const int i = (int)(blockIdx.x * 256u + threadIdx.x);
  if (i < n_elems) out[i] = bias[i & (D_FEAT - 1)];
}

// ---------------------------------------------------------------------------
// Kernel 3: edge scatter.  One wave32 per edge; each lane owns a float4 slice
// of the 128-float feature row (32 * 16B = 512B).  Edge metadata reads are
// forced wave-uniform (scalar loads); accumulation uses hardware
// global_atomic_add_f32.  support (51.2 MB) + out (51.2 MB) both fit in the
// 192 MB L2, so this phase runs at L2/atomic speed, not HBM speed.
// ---------------------------------------------------------------------------
__global__ __launch_bounds__(256) void gcn_edge_scatter(
    const float* __restrict__ S,        // support [N, 128]
    const int* __restrict__ erow,
    const int* __restrict__ ecol,
    const float* __restrict__ ew,
    float* __restrict__ out,            // [N, 128]
    int n_edges) {
  const int e = __builtin_amdgcn_readfirstlane(
      (int)(blockIdx.x * 8u + (threadIdx.x >> 5)));
  if (e >= n_edges) return;             // wave-uniform

  const int lane = threadIdx.x & 31;
  const int r = erow[e];
  const int c = ecol[e];
  const float wgt = ew[e];

  const v4f* __restrict__ src =
      (const v4f*)(S + (size_t)c * D_FEAT + lane * 4);
  v4f m = *src;
  m.x *= wgt; m.y *= wgt; m.z *= wgt; m.w *= wgt;

  float* __restrict__ dst = out + (size_t)r * D_FEAT + lane * 4;
  unsafeAtomicAdd(dst + 0, m.x);
  unsafeAtomicAdd(dst + 1, m.y);
  unsafeAtomicAdd(dst + 2, m.z);
  unsafeAtomicAdd(dst + 3, m.w);
}

// ---------------------------------------------------------------------------
// Launch: inputs = {X, edge_row, edge_col, edge_weight, W, bias}
// ---------------------------------------------------------------------------
extern "C" void kernel_launch(void* const* d_in, const int* in_sizes, int n_in,
                              void* d_out, int out_size, void* d_ws, size_t ws_size,
                              hipStream_t stream) {
  const float* X    = (const float*)d_in[0];
  const int*   erow = (const int*)d_in[1];
  const int*   ecol = (const int*)d_in[2];
  const float* ew   = (const float*)d_in[3];
  const float* W    = (const float*)d_in[4];
  const float* bias = (const float*)d_in[5];
  float* out = (float*)d_out;

  const int n_nodes = in_sizes[0] / D_FEAT;   // 100000
  const int n_edges = in_sizes[1];            // 3200000

  float* support = (float*)d_ws;              // n_nodes*128 floats = 51.2 MB

  // 1) support = X @ W (WMMA f32, W staged K-pair-interleaved in LDS)
  const int n_mtiles = (n_nodes + 15) / 16;   // 6250
  dim3 gemm_grid((n_mtiles + 7) / 8);
  gcn_gemm_wmma_f32<<<gemm_grid, 256, W_LDS_BYTES, stream>>>(X, W, support,
                                                             n_mtiles);

  // 2) out = broadcast(bias)
  const int n_elems = n_nodes * D_FEAT;
  gcn_init_out<<<(n_elems + 255) / 256, 256, 0, stream>>>(out, bias, n_elems);

  // 3) out[row] += w * support[col] over all edges (one wave per edge)
  dim3 sc_grid((n_edges + 7) / 8);
  gcn_edge_scatter<<<sc_grid, 256, 0, stream>>>(support, erow, ecol, ew, out,
                                                n_edges);
}